// ContentAttender_6597069767497
// MI455X (gfx1250) — compile-verified
//
#include <hip/hip_runtime.h>
#include <hip/hip_bf16.h>

// Problem constants (match reference): B=4, NQ=512, NK=512, D=128, H=32
constexpr int kB  = 4;
constexpr int kNQ = 512;
constexpr int kNK = 512;
constexpr int kD  = 128;
constexpr int kH  = 32;

typedef __attribute__((ext_vector_type(2))) float v2f;
typedef __attribute__((ext_vector_type(8))) float v8f;

// ---- CDNA5 transcendental helpers --------------------------------------
__device__ __forceinline__ float fast_tanh(float x) {
#if __has_builtin(__builtin_amdgcn_tanhf)
  return __builtin_amdgcn_tanhf(x);
#else
  float r;
  // v_tanh_f32 is a TRANS32 op; add one V_NOP to satisfy the multicycle
  // co-execution hazard (ISA 7.4) since the compiler can't classify raw asm.
  asm volatile("v_tanh_f32 %0, %1\n\tv_nop" : "=v"(r) : "v"(x));
  return r;
#endif
}

__device__ __forceinline__ float fast_exp(float x) {
#if __has_builtin(__builtin_amdgcn_exp2f)
  return __builtin_amdgcn_exp2f(x * 1.44269504088896340736f); // v_exp_f32 is 2^x
#else
  float r;
  float y = x * 1.44269504088896340736f;
  asm volatile("v_exp_f32 %0, %1\n\tv_nop" : "=v"(r) : "v"(y));
  return r;
#endif
}

// ---- Kernel 1: projections ---------------------------------------------
// qh[b,q,h]  = sum_d queries[b,q,d] * Wq[d,h]
// khT[b,h,k] = sum_d keys[b,k,d]    * Wk[d,h] + b1[h]   (bias folded, transposed)
__global__ __launch_bounds__(256) void k_project(
    const float* __restrict__ keys, const float* __restrict__ queries,
    const float* __restrict__ Wk, const float* __restrict__ Wq,
    const float* __restrict__ b1,
    float* __restrict__ qh, float* __restrict__ khT) {
  const int wave = threadIdx.x >> 5;          // 8 waves per block, one row each
  const int h    = threadIdx.x & 31;          // lane = hidden index
  const int row  = blockIdx.x * 8 + wave;     // 0 .. B*(NQ+NK)-1
  if (row < kB * kNQ) {
    const float* in = queries + (size_t)row * kD;   // wave-uniform row
    float acc = 0.f;
#pragma unroll 4
    for (int d = 0; d < kD; ++d)
      acc = fmaf(in[d], Wq[d * kH + h], acc);       // Wq load coalesced in h
    qh[(size_t)row * kH + h] = acc;
  } else {
    const int r = row - kB * kNQ;             // 0 .. B*NK-1
    const int b = r / kNK, k = r % kNK;
    const float* in = keys + (size_t)r * kD;
    float acc = b1[h];
#pragma unroll 4
    for (int d = 0; d < kD; ++d)
      acc = fmaf(in[d], Wk[d * kH + h], acc);
    khT[((size_t)(b * kH + h)) * kNK + k] = acc;    // transposed store
  }
}

// ---- Kernel 2: additive scores + softmax -------------------------------
// One block per (b,q); thread = k. logit = w2 . tanh(qh[b,q,:] + khT[b,:,k]).
// b2 is a uniform shift of all logits -> cancels in softmax (dropped).
__global__ __launch_bounds__(512) void k_scores(
    const float* __restrict__ qh, const float* __restrict__ khT,
    const float* __restrict__ w2, float* __restrict__ attn) {
  __shared__ float sqh[kH];
  __shared__ float sw2[kH];
  __shared__ float red[kNK];
  const int k  = threadIdx.x;                 // 0..511
  const int bq = blockIdx.x;                  // b*NQ + q
  const int b  = bq / kNQ;
  if (k < kH) {
    sqh[k] = qh[(size_t)bq * kH + k];
    sw2[k] = w2[k];
  }
  __syncthreads();

  const float* kt = khT + (size_t)b * kH * kNK + k;
  float acc = 0.f;
#pragma unroll
  for (int h = 0; h < kH; ++h)                // coalesced: lanes walk k
    acc = fmaf(sw2[h], fast_tanh(sqh[h] + kt[(size_t)h * kNK]), acc);

  // block-wide max
  red[k] = acc;
  __syncthreads();
  for (int s = kNK / 2; s > 0; s >>= 1) {
    if (k < s) red[k] = fmaxf(red[k], red[k + s]);
    __syncthreads();
  }
  const float m = red[0];
  __syncthreads();

  // exp + block-wide sum
  const float e = fast_exp(acc - m);
  red[k] = e;
  __syncthreads();
  for (int s = kNK / 2; s > 0; s >>= 1) {
    if (k < s) red[k] += red[k + s];
    __syncthreads();
  }
  attn[(size_t)bq * kNK + k] = e * (1.0f / red[0]);
}

// ---- Kernel 3: context = attn @ keys via V_WMMA_F32_16X16X4_F32 --------
// One wave per 16(q) x 16(d) tile; K-loop over NK in steps of 4.
// A 16x4 f32 layout: lanes 0-15 hold M, (vgpr0,lo)=K0 (vgpr1,lo)=K1
//                    (vgpr0,hi)=K2 (vgpr1,hi)=K3.
// B 4x16 f32 layout: row (fixed K) striped N across lanes; same K split.
// C 16x16 f32: vgpr i -> M=i (lanes 0-15) / M=8+i (lanes 16-31), N=lane%16.
__global__ __launch_bounds__(32) void k_context(
    const float* __restrict__ attn, const float* __restrict__ keys,
    float* __restrict__ out) {
  const int lane = threadIdx.x;
  const int half = lane >> 4;                 // K sub-block selector
  const int l    = lane & 15;                 // M (for A) / N (for B)
  const int DT = kD / 16, QT = kNQ / 16;
  const int t  = blockIdx.x;
  const int dT = t % DT;
  const int qT = (t / DT) % QT;
  const int b  = t / (DT * QT);
  const int q0 = qT * 16, d0 = dT * 16;

  const float* Arow = attn + ((size_t)b * kNQ + q0 + l) * kNK; // A row M=l
  const float* Bcol = keys + (size_t)b * kNK * kD + d0 + l;    // B col N=l

  v8f c = {};
  for (int kk = 0; kk < kNK; kk += 4) {
    const int ka = kk + half * 2;
    v2f a, bb;
    a.x  = Arow[ka];
    a.y  = Arow[ka + 1];
    bb.x = Bcol[(size_t)ka * kD];
    bb.y = Bcol[(size_t)(ka + 1) * kD];
    c = __builtin_amdgcn_wmma_f32_16x16x4_f32(
        /*neg_a=*/false, a, /*neg_b=*/false, bb,
        /*c_mod=*/(short)0, c, /*reuse_a=*/false, /*reuse_b=*/false);
  }

  float* o = out + ((size_t)b * kNQ + q0 + half * 8) * kD + d0 + l;
#pragma unroll
  for (int i = 0; i < 8; ++i) o[(size_t)i * kD] = c[i];
}

// ---- Launcher -----------------------------------------------------------
extern "C" void kernel_launch(void* const* d_in, const int* in_sizes, int n_in,
                              void* d_out, int out_size, void* d_ws, size_t ws_size,
                              hipStream_t stream) {
  const float* keys    = (const float*)d_in[0];
  const float* queries = (const float*)d_in[1];
  const float* Wk      = (const float*)d_in[2];
  const float* Wq      = (const float*)d_in[3];
  const float* b1      = (const float*)d_in[4];
  const float* w2      = (const float*)d_in[5];
  // d_in[6] = b2: uniform logit shift, cancels in softmax -> unused.
  float* out = (float*)d_out;

  // workspace layout (floats): qh | khT | attn  (~4.5 MB, L2-resident)
  float* qh   = (float*)d_ws;
  float* khT  = qh  + (size_t)kB * kNQ * kH;
  float* attn = khT + (size_t)kB * kH * kNK;

  k_project<<<(kB * (kNQ + kNK)) / 8, 256, 0, stream>>>(
      keys, queries, Wk, Wq, b1, qh, khT);
  k_scores<<<kB * kNQ, kNK, 0, stream>>>(qh, khT, w2, attn);
  k_context<<<kB * (kNQ / 16) * (kD / 16), 32, 0, stream>>>(attn, keys, out);
}